// EssentialMatrixEstimator_44220983279834
// MI455X (gfx1250) — compile-verified
//
#include <hip/hip_runtime.h>
#include <hip/hip_bf16.h>
#include <math.h>

#define N 2304
#define PSTRIDE 2305
#define NTILES 144            // N/16 column tiles
#define EPSF 1e-8f
#define INV_IRLS 100.0f       // 1/IRLS_SIGMA
#define INV_SS2 10000.0f      // 1/SAMPSON_SIGMA^2

typedef __attribute__((ext_vector_type(2))) float v2f;
typedef __attribute__((ext_vector_type(8))) float v8f;

// ---------------------------------------------------------------------------
// V_WMMA_F32_16X16X4_F32: D(16x16) = A(16x4) * B(4x16) + C.
// A: 2 VGPRs/lane: lane l holds A[l&15][k], k = (l>>4)*2 + {0,1}   (ISA 7.12.2)
// B: mirrored:    lane l holds B[(l>>4)*2 + {0,1}][l&15]
// C/D: 8 VGPRs:   lane l holds D[r + (l>>4)*8][l&15], r = 0..7
// ---------------------------------------------------------------------------
__device__ inline v8f wmma4(v2f a, v2f b, v8f c) {
  return __builtin_amdgcn_wmma_f32_16x16x4_f32(
      /*neg_a=*/false, a, /*neg_b=*/false, b,
      /*c_mod=*/(short)0, c, /*reuse_a=*/false, /*reuse_b=*/false);
}

__device__ inline void push3(float vv, float& m1, float& m2, float& m3) {
  if (vv > m1)      { m3 = m2; m2 = m1; m1 = vv; }
  else if (vv > m2) { m3 = m2; m2 = vv; }
  else if (vv > m3) { m3 = vv; }
}

// components of x1h x1h^T (upper triangle): (xx, xy, x, yy, y, 1)
__device__ inline float sym6(int comp, float x, float y) {
  switch (comp) {
    case 0: return x * x;
    case 1: return x * y;
    case 2: return x;
    case 3: return y * y;
    case 4: return y;
    case 5: return 1.0f;
    default: return 0.0f;
  }
}

__device__ inline float blockReduceSum256(float v, float* red) {
  int t = threadIdx.x;
  red[t] = v;
  __syncthreads();
  #pragma unroll
  for (int s = 128; s > 0; s >>= 1) {
    if (t < s) red[t] += red[t + s];
    __syncthreads();
  }
  float r = red[0];
  __syncthreads();
  return r;
}

// --------------------------- top-3 thresholds ------------------------------
__global__ __launch_bounds__(256) void rowTop3Kernel(const float* __restrict__ P,
                                                     float* __restrict__ thr_r) {
  __shared__ float buf[8][96];
  int wid = threadIdx.x >> 5, lane = threadIdx.x & 31;
  int row = blockIdx.x * 8 + wid;
  float m1 = -INFINITY, m2 = -INFINITY, m3 = -INFINITY;
  const float* rp = P + (size_t)row * PSTRIDE;
  for (int j = lane; j < N; j += 32) push3(rp[j], m1, m2, m3);
  buf[wid][lane * 3 + 0] = m1;
  buf[wid][lane * 3 + 1] = m2;
  buf[wid][lane * 3 + 2] = m3;
  __syncthreads();
  if (lane == 0) {
    float a = -INFINITY, b = -INFINITY, c = -INFINITY;
    for (int k = 0; k < 96; k++) push3(buf[wid][k], a, b, c);
    thr_r[row] = c;
  }
}

__global__ __launch_bounds__(256) void colTop3Kernel(const float* __restrict__ P,
                                                     float* __restrict__ thr_c) {
  int j = blockIdx.x * blockDim.x + threadIdx.x;
  if (j >= N) return;
  float m1 = -INFINITY, m2 = -INFINITY, m3 = -INFINITY;
  for (int i = 0; i < N; i++) push3(P[(size_t)i * PSTRIDE + j], m1, m2, m3);
  thr_c[j] = m3;
}

// --------------------------- per-solve prep --------------------------------
__global__ __launch_bounds__(256) void prepKernel(const float* __restrict__ pc,
                                                  const float* __restrict__ Ebuf,
                                                  float* __restrict__ Ex1,
                                                  float* __restrict__ e1n,
                                                  float* __restrict__ e2n) {
  int i = blockIdx.x * blockDim.x + threadIdx.x;
  if (i >= N) return;
  float x = pc[2 * i], y = pc[2 * i + 1];
  float E00 = Ebuf[0], E01 = Ebuf[1], E02 = Ebuf[2];
  float E10 = Ebuf[3], E11 = Ebuf[4], E12 = Ebuf[5];
  float E20 = Ebuf[6], E21 = Ebuf[7], E22 = Ebuf[8];
  float a0 = E00 * x + E01 * y + E02;
  float a1 = E10 * x + E11 * y + E12;
  float a2 = E20 * x + E21 * y + E22;
  Ex1[i * 4 + 0] = a0; Ex1[i * 4 + 1] = a1;
  Ex1[i * 4 + 2] = a2; Ex1[i * 4 + 3] = 0.0f;     // K=4 pad for WMMA
  e1n[i] = a0 * a0 + a1 * a1;
  float b0 = E00 * x + E10 * y + E20;              // E^T x2
  float b1 = E01 * x + E11 * y + E21;
  e2n[i] = b0 * b0 + b1 * b1;
}

// ------------------- main fused pass: w, sums, S_raw -----------------------
// One block per 16-column tile; 4 waves split the 2304 rows (36 chunks each).
// mode 0: w = w0 ; mode 1: IRLS ; mode 2: Sampson.
__global__ __launch_bounds__(128) void pass1Kernel(
    const float* __restrict__ P, const float* __restrict__ thr_r,
    const float* __restrict__ thr_c, const float* __restrict__ pc,
    const float* __restrict__ Ex1, const float* __restrict__ e1n,
    const float* __restrict__ e2n, int mode,
    float* __restrict__ rsumPart, float* __restrict__ csum,
    float* __restrict__ Sraw) {
  __shared__ float wtile[4][16][17];
  __shared__ float Spart[4][16][6];
  __shared__ float csLDS[4][16];

  const int jt = blockIdx.x, j0 = jt * 16;
  const int wid = threadIdx.x >> 5, l = threadIdx.x & 31;
  const int lm = l & 15, half = l >> 4, kb0 = half * 2;

  const float tc = thr_c[j0 + lm];
  const float jx = pc[2 * (j0 + lm)], jy = pc[2 * (j0 + lm) + 1];
  // B for residual WMMA: rows of x2h^T = (x, y, 1, 0)
  v2f rB;
  rB.x = (half == 0) ? jx : 1.0f;
  rB.y = (half == 0) ? jy : 0.0f;
  const float e2v = (mode == 2) ? e2n[j0 + lm] : 0.0f;

  v8f sacc = {0.f, 0.f, 0.f, 0.f, 0.f, 0.f, 0.f, 0.f};
  float csacc = 0.0f;

  for (int i0 = wid * 16; i0 < N; i0 += 64) {
    // residual tile r(16i x 16j) = Ex1(16x4) * x2h^T(4x16) via WMMA
    v8f rd = {0.f, 0.f, 0.f, 0.f, 0.f, 0.f, 0.f, 0.f};
    if (mode != 0) {
      const int ia = i0 + lm;
      v2f rA;
      rA.x = Ex1[ia * 4 + kb0];
      rA.y = Ex1[ia * 4 + kb0 + 1];
      rd = wmma4(rA, rB, rd);
    }
    // fused w0 (top-k gate on P) + robust reweighting, in D layout
    #pragma unroll
    for (int r = 0; r < 8; r++) {
      const int m = r + half * 8, gi = i0 + m;
      float Pv = P[(size_t)gi * PSTRIDE + j0 + lm];
      float w0v = (Pv >= thr_r[gi] && Pv >= tc) ? Pv : 0.0f;
      float w;
      if (mode == 0) {
        w = w0v;
      } else if (mode == 1) {
        float t = rd[r] * INV_IRLS;
        w = w0v / (1.0f + t * t);
      } else {
        float dnm = e1n[gi] + e2v + EPSF;
        float d = rd[r] * rd[r] / dnm;
        w = w0v / (1.0f + d * INV_SS2);
      }
      csacc += w;
      wtile[wid][m][lm] = w;
    }
    __syncthreads();
    // row-sum partials (Hartley weights for pts1)
    if (half == 0) {
      float rs = 0.0f;
      #pragma unroll
      for (int nn = 0; nn < 16; nn++) rs += wtile[wid][lm][nn];
      rsumPart[(size_t)jt * N + i0 + lm] = rs;
    }
    // S_raw(16j x 6) += w^T(16x4) * X1sym(4x16) via 4 WMMAs over the 16 rows
    #pragma unroll
    for (int c4 = 0; c4 < 4; c4++) {
      v2f a2;
      a2.x = wtile[wid][4 * c4 + kb0][lm];
      a2.y = wtile[wid][4 * c4 + kb0 + 1][lm];
      const int i1 = i0 + 4 * c4 + kb0, i2 = i1 + 1;
      v2f b2;
      b2.x = sym6(lm, pc[2 * i1], pc[2 * i1 + 1]);
      b2.y = sym6(lm, pc[2 * i2], pc[2 * i2 + 1]);
      sacc = wmma4(a2, b2, sacc);
    }
    __syncthreads();
  }
  // combine 2 lane-halves + 4 waves
  float cs = csacc + __shfl_xor(csacc, 16);
  if (half == 0) csLDS[wid][lm] = cs;
  if (lm < 6) {
    #pragma unroll
    for (int r = 0; r < 8; r++) Spart[wid][r + half * 8][lm] = sacc[r];
  }
  __syncthreads();
  for (int idx = threadIdx.x; idx < 96; idx += 128) {
    int jl = idx / 6, cmp = idx % 6;
    Sraw[(size_t)(j0 + jl) * 6 + cmp] =
        Spart[0][jl][cmp] + Spart[1][jl][cmp] + Spart[2][jl][cmp] + Spart[3][jl][cmp];
  }
  if (threadIdx.x < 16) {
    int t = threadIdx.x;
    csum[j0 + t] = csLDS[0][t] + csLDS[1][t] + csLDS[2][t] + csLDS[3][t];
  }
}

__global__ __launch_bounds__(256) void reduceRsumKernel(const float* __restrict__ rsumPart,
                                                        float* __restrict__ rsum) {
  int i = blockIdx.x * blockDim.x + threadIdx.x;
  if (i >= N) return;
  float s = 0.0f;
  for (int b = 0; b < NTILES; b++) s += rsumPart[(size_t)b * N + i];
  rsum[i] = s;
}

// ------------------------- Hartley normalization ---------------------------
__global__ __launch_bounds__(256) void hartleyKernel(const float* __restrict__ rsum,
                                                     const float* __restrict__ csum,
                                                     const float* __restrict__ pc,
                                                     float* __restrict__ Tbuf) {
  __shared__ float red[256];
  for (int k = 0; k < 2; k++) {
    const float* w = k ? csum : rsum;
    float sw = 0.f, sx = 0.f, sy = 0.f;
    for (int i = threadIdx.x; i < N; i += 256) {
      float wi = w[i];
      sw += wi;
      sx += wi * pc[2 * i];
      sy += wi * pc[2 * i + 1];
    }
    sw = blockReduceSum256(sw, red);
    sx = blockReduceSum256(sx, red);
    sy = blockReduceSum256(sy, red);
    float ws = sw + EPSF;
    float cx = sx / ws, cy = sy / ws;
    float sd = 0.f;
    for (int i = threadIdx.x; i < N; i += 256) {
      float dx = pc[2 * i] - cx, dy = pc[2 * i + 1] - cy;
      sd += w[i] * (dx * dx + dy * dy);
    }
    sd = blockReduceSum256(sd, red);
    float md = sqrtf(sd / ws + EPSF);
    float s = 1.41421356237f / (md + EPSF);
    if (threadIdx.x == 0) {
      Tbuf[k * 3 + 0] = s;
      Tbuf[k * 3 + 1] = cx;
      Tbuf[k * 3 + 2] = cy;
    }
    __syncthreads();
  }
}

// ------------- M9 assembly: M9 = sum_j Y2_j (x) (T1 S_j T1^T) --------------
__global__ __launch_bounds__(256) void m9Kernel(const float* __restrict__ Sraw,
                                                const float* __restrict__ pc,
                                                const float* __restrict__ Tbuf,
                                                float* __restrict__ P36) {
  __shared__ float red[256];
  float s1 = Tbuf[0], c1x = Tbuf[1], c1y = Tbuf[2];
  float s2 = Tbuf[3], c2x = Tbuf[4], c2y = Tbuf[5];
  float acc[36];
  #pragma unroll
  for (int e = 0; e < 36; e++) acc[e] = 0.0f;

  for (int j = threadIdx.x; j < N; j += 256) {
    float y0 = s2 * (pc[2 * j] - c2x);
    float y1 = s2 * (pc[2 * j + 1] - c2y);
    float Y[6] = {y0 * y0, y0 * y1, y0, y1 * y1, y1, 1.0f};
    float S00 = Sraw[j * 6 + 0], S01 = Sraw[j * 6 + 1], S02 = Sraw[j * 6 + 2];
    float S11 = Sraw[j * 6 + 3], S12 = Sraw[j * 6 + 4], S22 = Sraw[j * 6 + 5];
    // M' = T1 @ S  (T1 = [[s1,0,-s1*c1x],[0,s1,-s1*c1y],[0,0,1]])
    float a00 = s1 * (S00 - c1x * S02), a01 = s1 * (S01 - c1x * S12), a02 = s1 * (S02 - c1x * S22);
    float a11 = s1 * (S11 - c1y * S12), a12 = s1 * (S12 - c1y * S22);
    float a22 = S22;
    // Nrm = M' @ T1^T (symmetric)
    float Nv[6];
    Nv[0] = s1 * (a00 - c1x * a02);
    Nv[1] = s1 * (a01 - c1y * a02);
    Nv[2] = a02;
    Nv[3] = s1 * (a11 - c1y * a12);
    Nv[4] = a12;
    Nv[5] = a22;
    #pragma unroll
    for (int u = 0; u < 6; u++)
      #pragma unroll
      for (int v = 0; v < 6; v++) acc[6 * u + v] += Y[u] * Nv[v];
  }
  for (int e = 0; e < 36; e++) {
    float r = blockReduceSum256(acc[e], red);
    if (threadIdx.x == 0) P36[e] = r;
  }
}

// ------------- tiny linear algebra: eig9, unnormalize, project_E -----------
__global__ void solveSmallKernel(const float* __restrict__ P36,
                                 const float* __restrict__ Tbuf,
                                 float* __restrict__ Ebuf,
                                 float* __restrict__ outE, int writeOut) {
  if (threadIdx.x != 0 || blockIdx.x != 0) return;
  const int sidx[3][3] = {{0, 1, 2}, {1, 3, 4}, {2, 4, 5}};
  float M9[9][9];
  for (int a = 0; a < 3; a++)
    for (int b = 0; b < 3; b++)
      for (int c = 0; c < 3; c++)
        for (int d = 0; d < 3; d++)
          M9[a * 3 + b][c * 3 + d] = P36[sidx[a][c] * 6 + sidx[b][d]];
  float lam = 0.0f;
  for (int i = 0; i < 9; i++) lam += M9[i][i];
  float Ms[9][9];
  for (int i = 0; i < 9; i++)
    for (int j = 0; j < 9; j++) Ms[i][j] = ((i == j) ? lam : 0.0f) - M9[i][j];
  float v[9];
  for (int i = 0; i < 9; i++) v[i] = 1.0f / 3.0f;
  for (int it = 0; it < 30; it++) {
    float u[9], nrm = 0.0f;
    for (int i = 0; i < 9; i++) {
      float s = 0.0f;
      for (int j = 0; j < 9; j++) s += Ms[i][j] * v[j];
      u[i] = s;
      nrm += s * s;
    }
    nrm = sqrtf(nrm) + EPSF;
    for (int i = 0; i < 9; i++) v[i] = u[i] / nrm;
  }
  float s1 = Tbuf[0], c1x = Tbuf[1], c1y = Tbuf[2];
  float s2 = Tbuf[3], c2x = Tbuf[4], c2y = Tbuf[5];
  float F[3][3];
  for (int a = 0; a < 3; a++)
    for (int b = 0; b < 3; b++) F[a][b] = v[a * 3 + b];
  float G[3][3];
  for (int b = 0; b < 3; b++) {
    G[0][b] = s2 * F[0][b];
    G[1][b] = s2 * F[1][b];
    G[2][b] = -s2 * c2x * F[0][b] - s2 * c2y * F[1][b] + F[2][b];
  }
  float E0[3][3];
  for (int r = 0; r < 3; r++) {
    E0[r][0] = s1 * G[r][0];
    E0[r][1] = s1 * G[r][1];
    E0[r][2] = -s1 * c1x * G[r][0] - s1 * c1y * G[r][1] + G[r][2];
  }
  // ---- project_E ----
  float B3[3][3];
  for (int i = 0; i < 3; i++)
    for (int j = 0; j < 3; j++) {
      float s = 0.0f;
      for (int k = 0; k < 3; k++) s += E0[k][i] * E0[k][j];
      B3[i][j] = s;
    }
  float lam3 = B3[0][0] + B3[1][1] + B3[2][2];
  float Ms3[3][3];
  for (int i = 0; i < 3; i++)
    for (int j = 0; j < 3; j++) Ms3[i][j] = ((i == j) ? lam3 : 0.0f) - B3[i][j];
  const float iv3 = 0.57735026919f;
  float v1[3] = {iv3, iv3, iv3}, v3a[3] = {iv3, iv3, iv3};
  for (int it = 0; it < 10; it++) {
    float u[3], nrm = 0.0f;
    for (int i = 0; i < 3; i++) {
      u[i] = B3[i][0] * v1[0] + B3[i][1] * v1[1] + B3[i][2] * v1[2];
      nrm += u[i] * u[i];
    }
    nrm = sqrtf(nrm) + EPSF;
    for (int i = 0; i < 3; i++) v1[i] = u[i] / nrm;
  }
  for (int it = 0; it < 10; it++) {
    float u[3], nrm = 0.0f;
    for (int i = 0; i < 3; i++) {
      u[i] = Ms3[i][0] * v3a[0] + Ms3[i][1] * v3a[1] + Ms3[i][2] * v3a[2];
      nrm += u[i] * u[i];
    }
    nrm = sqrtf(nrm) + EPSF;
    for (int i = 0; i < 3; i++) v3a[i] = u[i] / nrm;
  }
  float v2c[3] = {v3a[1] * v1[2] - v3a[2] * v1[1],
                  v3a[2] * v1[0] - v3a[0] * v1[2],
                  v3a[0] * v1[1] - v3a[1] * v1[0]};
  float n2 = sqrtf(v2c[0] * v2c[0] + v2c[1] * v2c[1] + v2c[2] * v2c[2]) + EPSF;
  v2c[0] /= n2; v2c[1] /= n2; v2c[2] /= n2;
  float a[3], bb[3];
  for (int r = 0; r < 3; r++) {
    a[r]  = E0[r][0] * v1[0]  + E0[r][1] * v1[1]  + E0[r][2] * v1[2];
    bb[r] = E0[r][0] * v2c[0] + E0[r][1] * v2c[1] + E0[r][2] * v2c[2];
  }
  float sa = sqrtf(a[0] * a[0] + a[1] * a[1] + a[2] * a[2]);
  float sb = sqrtf(bb[0] * bb[0] + bb[1] * bb[1] + bb[2] * bb[2]);
  float savg = 0.5f * (sa + sb);
  float u1[3], u2[3];
  for (int r = 0; r < 3; r++) {
    u1[r] = a[r] / (sa + EPSF);
    u2[r] = bb[r] / (sb + EPSF);
  }
  // E = U diag(savg,savg,0) V^T = savg*(u1 v1^T + u2 v2^T); the det-sign
  // flips only touch column 2 of U/V, which is killed by the zero singular value.
  for (int r = 0; r < 3; r++)
    for (int c = 0; c < 3; c++) {
      float e = savg * (u1[r] * v1[c] + u2[r] * v2c[c]);
      Ebuf[r * 3 + c] = e;
      if (writeOut) outE[r * 3 + c] = e;
    }
}

// ---------------------------------------------------------------------------
extern "C" void kernel_launch(void* const* d_in, const int* in_sizes, int n_in,
                              void* d_out, int out_size, void* d_ws, size_t ws_size,
                              hipStream_t stream) {
  (void)in_sizes; (void)n_in; (void)out_size; (void)ws_size;
  const float* P  = (const float*)d_in[0];   // (2305, 2305) f32
  const float* pc = (const float*)d_in[1];   // (2304, 2) f32

  float* ws = (float*)d_ws;
  float* thr_r    = ws;                    // N
  float* thr_c    = thr_r + N;             // N
  float* Ex1      = thr_c + N;             // 4N (padded to 4 for WMMA A)
  float* e1n      = Ex1 + 4 * N;           // N
  float* e2n      = e1n + N;               // N
  float* rsum     = e2n + N;               // N
  float* csum     = rsum + N;              // N
  float* Sraw     = csum + N;              // 6N
  float* rsumPart = Sraw + 6 * N;          // 144*N
  float* Tbuf     = rsumPart + (size_t)NTILES * N;  // 8
  float* P36      = Tbuf + 8;              // 36
  float* Ebuf     = P36 + 36;              // 16
  // total ~160*N + 60 floats ~ 1.5 MB

  rowTop3Kernel<<<N / 8, 256, 0, stream>>>(P, thr_r);
  colTop3Kernel<<<N / 256, 256, 0, stream>>>(P, thr_c);

  for (int s = 0; s < 9; s++) {
    int mode = (s == 0) ? 0 : ((s <= 5) ? 1 : 2);  // 1 init + 5 IRLS + 3 Sampson
    if (mode != 0)
      prepKernel<<<N / 256, 256, 0, stream>>>(pc, Ebuf, Ex1, e1n, e2n);
    pass1Kernel<<<NTILES, 128, 0, stream>>>(P, thr_r, thr_c, pc, Ex1, e1n, e2n,
                                            mode, rsumPart, csum, Sraw);
    reduceRsumKernel<<<N / 256, 256, 0, stream>>>(rsumPart, rsum);
    hartleyKernel<<<1, 256, 0, stream>>>(rsum, csum, pc, Tbuf);
    m9Kernel<<<1, 256, 0, stream>>>(Sraw, pc, Tbuf, P36);
    solveSmallKernel<<<1, 32, 0, stream>>>(P36, Tbuf, Ebuf, (float*)d_out,
                                           (s == 8) ? 1 : 0);
  }
}